// EmbedTrackLoss_33354716021219
// MI455X (gfx1250) — compile-verified
//
#include <hip/hip_runtime.h>
#include <hip/hip_bf16.h>

// ---------------------------------------------------------------------------
// EmbedTrack loss for MI455X (gfx1250, wave32).
// Per-instance masked reductions are done with V_WMMA_F32_16X16X4_F32:
//   D[16 inst x 16 feat] += A[16 inst x 4 px one-hot] * B[4 px x 16 feat]
// Lovasz hinge: pack (orderable float bits | label) into one u32 per pixel,
// bitonic-sort 96 arrays of 262144 keys (LDS-fused inner stages), then a
// block-scan computes the Jaccard-gradient dot product.
// ---------------------------------------------------------------------------

#define HH 512
#define WW 512
#define NPIX (HH * WW)      // 262144
#define NB 4
#define NI 16
#define TILE 2048
#define NSEG (NB * NI)      // 64 seg arrays
#define NARR (NSEG + 2 * NI) // + track arrays for b<2 -> 96

typedef __attribute__((ext_vector_type(2))) float v2f;
typedef __attribute__((ext_vector_type(8))) float v8f;

__device__ inline float sigmf(float x) { return 1.0f / (1.0f + __expf(-x)); }

// Monotonic float <-> orderable-uint maps (ascending).
__device__ inline unsigned f2ord(float f) {
    unsigned u = __float_as_uint(f);
    return (u & 0x80000000u) ? ~u : (u | 0x80000000u);
}
__device__ inline float ord2f(unsigned o) {
    unsigned u = (o & 0x80000000u) ? (o & 0x7fffffffu) : ~o;
    return __uint_as_float(u);
}

// ---------------------------------------------------------------------------
__global__ void k_zero(float* p, int n) {
    int t = blockIdx.x * blockDim.x + threadIdx.x;
    if (t < n) p[t] = 0.0f;
}

// ---------------------------------------------------------------------------
// K1: per-instance feature sums via WMMA f32 16x16x4.
// stats[b][inst][feat], feat: 0=1,1=ym,2=xm,3=s0,4=s1,5=s0^2,6=s1^2,
// 7=cent,8=cent*ym,9=cent*xm,10=cent*(ym-off0/1024),11=cent*(xm-off1/1024)
__global__ __launch_bounds__(256) void k_stats(
    const float* __restrict__ seg, const float* __restrict__ offs,
    const int* __restrict__ inst, const int* __restrict__ lab,
    const int* __restrict__ cent, float* __restrict__ stats,
    float* __restrict__ bg) {
    const int b = blockIdx.y;
    const int wave = threadIdx.x >> 5;
    const int lane = threadIdx.x & 31;

    __shared__ float ldsF[8][32][16];
    __shared__ int   ldsI[8][32];
    __shared__ float red[256];

    const float* p2 = seg + ((size_t)b * 5 + 2) * NPIX;
    const float* p3 = seg + ((size_t)b * 5 + 3) * NPIX;
    const float* p4 = seg + ((size_t)b * 5 + 4) * NPIX;
    const float* o0 = offs + ((size_t)b * 2 + 0) * NPIX;
    const float* o1 = offs + ((size_t)b * 2 + 1) * NPIX;
    const int* ib = inst + (size_t)b * NPIX;
    const int* lb = lab + (size_t)b * NPIX;
    const int* cb = cent + (size_t)b * NPIX;

    const int pxPerBlock = NPIX / 64;            // grid.x == 64
    const int blockBase = blockIdx.x * pxPerBlock;
    const int chunks = pxPerBlock / (8 * 32);    // 16

    v8f acc = {0.f, 0.f, 0.f, 0.f, 0.f, 0.f, 0.f, 0.f};
    float bgc = 0.f, bgs = 0.f;

    const int half = lane >> 4;
    const int l16 = lane & 15;

    for (int c = 0; c < chunks; ++c) {
        int p = blockBase + (c * 8 + wave) * 32 + lane;
        int y = p >> 9, x = p & 511;
        float ym = (float)y * (1.0f / 1023.0f);
        float xm = (float)x * (1.0f / 1023.0f);
        float s0 = sigmf(p2[p]);
        float s1 = sigmf(p3[p]);
        float cm = (cb[p] != 0) ? 1.0f : 0.0f;
        float f[16];
        f[0] = 1.0f; f[1] = ym; f[2] = xm; f[3] = s0; f[4] = s1;
        f[5] = s0 * s0; f[6] = s1 * s1; f[7] = cm;
        f[8] = cm * ym; f[9] = cm * xm;
        f[10] = cm * (ym - o0[p] * (1.0f / 1024.0f));
        f[11] = cm * (xm - o1[p] * (1.0f / 1024.0f));
        f[12] = 0.f; f[13] = 0.f; f[14] = 0.f; f[15] = 0.f;
#pragma unroll
        for (int ff = 0; ff < 16; ++ff) ldsF[wave][lane][ff] = f[ff];
        ldsI[wave][lane] = ib[p];

        if (lb[p] == 0) { bgc += 1.0f; float sd = sigmf(p4[p]); bgs += sd * sd; }
        __syncthreads();

#pragma unroll
        for (int s = 0; s < 8; ++s) {
            int q0 = 4 * s + 2 * half;  // pixel for K=0 (half0) / K=2 (half1)
            int q1 = q0 + 1;            // pixel for K=1 / K=3
            int i0 = ldsI[wave][q0];
            int i1 = ldsI[wave][q1];
            v2f av, bv;
            av.x = (i0 == l16 + 1) ? 1.0f : 0.0f;
            av.y = (i1 == l16 + 1) ? 1.0f : 0.0f;
            bv.x = ldsF[wave][q0][l16];
            bv.y = ldsF[wave][q1][l16];
            acc = __builtin_amdgcn_wmma_f32_16x16x4_f32(
                false, av, false, bv, (short)0, acc, false, false);
        }
        __syncthreads();
    }

    // Scatter D (16x16) into global stats with atomics.
    int Mb = (lane < 16) ? 0 : 8;
#pragma unroll
    for (int r = 0; r < 8; ++r)
        atomicAdd(&stats[((size_t)b * NI + (Mb + r)) * 16 + l16], acc[r]);

    // Background seed reduction.
    red[threadIdx.x] = bgc; __syncthreads();
    for (int off = 128; off > 0; off >>= 1) {
        if (threadIdx.x < off) red[threadIdx.x] += red[threadIdx.x + off];
        __syncthreads();
    }
    if (threadIdx.x == 0) atomicAdd(&bg[b * 2 + 0], red[0]);
    __syncthreads();
    red[threadIdx.x] = bgs; __syncthreads();
    for (int off = 128; off > 0; off >>= 1) {
        if (threadIdx.x < off) red[threadIdx.x] += red[threadIdx.x + off];
        __syncthreads();
    }
    if (threadIdx.x == 0) atomicAdd(&bg[b * 2 + 1], red[0]);
}

// ---------------------------------------------------------------------------
// K2: finalize per-(b,inst) params: center, s=exp(10*s_mean), gt_prev, var,
// valid, gts, npix_clamped.  params stride 12 floats.
__global__ void k_finalize(const float* __restrict__ stats, float* __restrict__ params) {
    int t = threadIdx.x;
    if (t >= NB * NI) return;
    const float* S = stats + (size_t)t * 16;
    float raw = S[0];
    float npix = fmaxf(raw, 1.0f);
    float ccnt = S[7];
    float cy = (ccnt == 1.0f) ? S[8] : S[1] / npix;
    float cx = (ccnt == 1.0f) ? S[9] : S[2] / npix;
    float sm0 = S[3] / npix, sm1 = S[4] / npix;
    float var = (S[5] - S[3] * sm0 + S[6] - S[4] * sm1) / (2.0f * npix);
    float* P = params + (size_t)t * 12;
    P[0] = cy; P[1] = cx;
    P[2] = __expf(10.0f * sm0);
    P[3] = __expf(10.0f * sm1);
    P[4] = S[10]; P[5] = S[11];           // gt_prev (y,x)
    P[6] = (ccnt > 0.0f) ? 1.0f : 0.0f;   // valid
    P[7] = raw;                           // gts (sum of labels)
    P[8] = npix;                          // clamped
    P[9] = var;
    P[10] = 0.f; P[11] = 0.f;
}

// ---------------------------------------------------------------------------
// K3: per-pixel errors for seg (all b) and track (b<2); sortable keys with
// label in LSB, stored bit-inverted so ascending sort == descending errors.
__global__ __launch_bounds__(256) void k_errors(
    const float* __restrict__ seg, const float* __restrict__ offp,
    const int* __restrict__ inst, const float* __restrict__ params,
    unsigned* __restrict__ errs, float* __restrict__ seedSum) {
    const int b = blockIdx.y;
    __shared__ float P[NI][12];
    __shared__ float sbin[NI];
    int t = threadIdx.x;
    if (t < NI * 12) ((float*)P)[t] = params[(size_t)(b * NI) * 12 + t];
    if (t < NI) sbin[t] = 0.0f;
    __syncthreads();

    const float* p0 = seg + ((size_t)b * 5 + 0) * NPIX;
    const float* p1 = seg + ((size_t)b * 5 + 1) * NPIX;
    const float* p4 = seg + ((size_t)b * 5 + 4) * NPIX;
    const float* q0 = offp + ((size_t)b * 2 + 0) * NPIX;
    const float* q1 = offp + ((size_t)b * 2 + 1) * NPIX;
    const int* ib = inst + (size_t)b * NPIX;
    const bool track = (b < NB / 2);
    const int stride = gridDim.x * blockDim.x;

    for (int p = blockIdx.x * blockDim.x + t; p < NPIX; p += stride) {
        int y = p >> 9, x = p & 511;
        float ym = (float)y * (1.0f / 1023.0f);
        float xm = (float)x * (1.0f / 1023.0f);
        float ey = tanhf(p0[p]) + ym;
        float ex = tanhf(p1[p]) + xm;
        float ty = ym - tanhf(q0[p]);
        float tx = xm - tanhf(q1[p]);
        float seed = sigmf(p4[p]);
        int iid = ib[p];
#pragma unroll 4
        for (int i = 0; i < NI; ++i) {
            float s0 = P[i][2], s1 = P[i][3];
            float dy = ey - P[i][0], dx = ex - P[i][1];
            float d = __expf(-(dy * dy * s0 + dx * dx * s1));
            unsigned mlab = (iid == i + 1) ? 1u : 0u;
            float sign = 2.0f * (float)mlab - 1.0f;
            float err = 1.0f - (2.0f * d - 1.0f) * sign;
            unsigned key = (f2ord(err) & ~1u) | mlab;
            errs[(size_t)(b * NI + i) * NPIX + p] = ~key;
            if (mlab) { float dd = seed - d; atomicAdd(&sbin[i], dd * dd); }
            if (track) {
                float uy = ty - P[i][4], ux = tx - P[i][5];
                float dt = __expf(-(uy * uy * s0 + ux * ux * s1));
                float errt = 1.0f - (2.0f * dt - 1.0f) * sign;
                unsigned kt = (f2ord(errt) & ~1u) | mlab;
                errs[(size_t)(NSEG + b * NI + i) * NPIX + p] = ~kt;
            }
        }
    }
    __syncthreads();
    if (t < NI) atomicAdd(&seedSum[b * NI + t], sbin[t]);
}

// ---------------------------------------------------------------------------
// Bitonic sort (ascending on stored values) over 96 arrays of NPIX u32.
__device__ inline void cswap(unsigned& a, unsigned& b, bool up) {
    if ((a > b) == up) { unsigned t = a; a = b; b = t; }
}

__global__ __launch_bounds__(1024) void k_sort_local(unsigned* __restrict__ data) {
    __shared__ unsigned sh[TILE];
    unsigned* arr = data + (size_t)blockIdx.y * NPIX + (size_t)blockIdx.x * TILE;
    int t = threadIdx.x;
    sh[t] = arr[t];
    sh[t + 1024] = arr[t + 1024];
    __syncthreads();
    unsigned gbase = blockIdx.x * TILE;
    for (unsigned k = 2; k <= TILE; k <<= 1) {
        for (unsigned j = k >> 1; j > 0; j >>= 1) {
            unsigned i = (((unsigned)t & ~(j - 1)) << 1) | ((unsigned)t & (j - 1));
            bool up = (((gbase + i) & k) == 0);
            cswap(sh[i], sh[i + j], up);
            __syncthreads();
        }
    }
    arr[t] = sh[t];
    arr[t + 1024] = sh[t + 1024];
}

__global__ __launch_bounds__(256) void k_sort_global(unsigned* __restrict__ data,
                                                     unsigned k, unsigned j) {
    unsigned t = blockIdx.x * 256u + threadIdx.x;   // 0 .. NPIX/2-1
    unsigned i = ((t & ~(j - 1)) << 1) | (t & (j - 1));
    unsigned* arr = data + (size_t)blockIdx.y * NPIX;
    bool up = ((i & k) == 0);
    unsigned a = arr[i], b = arr[i + j];
    if ((a > b) == up) { arr[i] = b; arr[i + j] = a; }
}

__global__ __launch_bounds__(1024) void k_merge_local(unsigned* __restrict__ data,
                                                      unsigned k) {
    __shared__ unsigned sh[TILE];
    unsigned* arr = data + (size_t)blockIdx.y * NPIX + (size_t)blockIdx.x * TILE;
    int t = threadIdx.x;
    sh[t] = arr[t];
    sh[t + 1024] = arr[t + 1024];
    __syncthreads();
    unsigned gbase = blockIdx.x * TILE;
    for (unsigned j = 1024; j > 0; j >>= 1) {
        unsigned i = (((unsigned)t & ~(j - 1)) << 1) | ((unsigned)t & (j - 1));
        bool up = (((gbase + i) & k) == 0);
        cswap(sh[i], sh[i + j], up);
        __syncthreads();
    }
    arr[t] = sh[t];
    arr[t + 1024] = sh[t + 1024];
}

// ---------------------------------------------------------------------------
// K5: Lovasz hinge over sorted (descending) errors via running label scan.
__global__ __launch_bounds__(256) void k_lovasz(const unsigned* __restrict__ errs,
                                                const float* __restrict__ params,
                                                float* __restrict__ results) {
    const int a = blockIdx.x;                       // 0..95
    const int bi = (a < NSEG) ? a : (a - NSEG);     // params index b*16+i
    const float gts = params[(size_t)bi * 12 + 7];
    const unsigned* arr = errs + (size_t)a * NPIX;
    __shared__ float scan[256];
    int t = threadIdx.x;
    float carry = 0.0f;
    float accLocal = 0.0f;

    for (int c = 0; c < NPIX / 256; ++c) {
        unsigned v = ~arr[c * 256 + t];
        float label = (float)(v & 1u);
        float err = ord2f(v & ~1u);
        scan[t] = label;
        __syncthreads();
        for (int off = 1; off < 256; off <<= 1) {
            float add = (t >= off) ? scan[t - off] : 0.0f;
            __syncthreads();
            scan[t] += add;
            __syncthreads();
        }
        float ci = carry + scan[t];
        float cim1 = ci - label;
        float pos = (float)(c * 256 + t + 1);
        float jac = 1.0f - (gts - ci) / (gts + (pos - ci));
        float contrib;
        if (c == 0 && t == 0) {
            contrib = fmaxf(err, 0.0f) * jac;
        } else {
            float jacm = 1.0f - (gts - cim1) / (gts + (pos - 1.0f - cim1));
            contrib = fmaxf(err, 0.0f) * (jac - jacm);
        }
        accLocal += contrib;
        float chunkTotal = scan[255];
        __syncthreads();
        carry += chunkTotal;
    }

    scan[t] = accLocal;
    __syncthreads();
    for (int off = 128; off > 0; off >>= 1) {
        if (t < off) scan[t] += scan[t + off];
        __syncthreads();
    }
    if (t == 0) results[a] = scan[0];
}

// ---------------------------------------------------------------------------
// K6: combine everything into the scalar loss.
__global__ void k_final(const float* __restrict__ params,
                        const float* __restrict__ seedSum,
                        const float* __restrict__ bg,
                        const float* __restrict__ results,
                        float* __restrict__ out) {
    if (threadIdx.x != 0 || blockIdx.x != 0) return;
    float loss = 0.0f, tsum = 0.0f, tcnt = 0.0f;
    for (int b = 0; b < NB; ++b) {
        float segs = 0.f, vars = 0.f, seeds = 0.f;
        for (int i = 0; i < NI; ++i) {
            const float* P = params + (size_t)(b * NI + i) * 12;
            float valid = P[6];
            segs += results[b * NI + i] * valid;
            vars += P[9] * valid;
            seeds += (seedSum[b * NI + i] / P[8]) * valid;   // FG_W = 1
        }
        float bgc = bg[b * 2 + 0], bgs = bg[b * 2 + 1];
        float seed_bg = (bgc > 0.0f) ? bgs / fmaxf(bgc, 1.0f) : 0.0f;
        loss += 1.0f * segs + 10.0f * vars + 1.0f * (seed_bg + seeds);
        if (b < NB / 2) {
            for (int i = 0; i < NI; ++i) {
                const float* P = params + (size_t)(b * NI + i) * 12;
                tsum += results[NSEG + b * NI + i] * P[6];
                tcnt += P[6];
            }
        }
    }
    float track = (tcnt > 0.0f) ? tsum / fmaxf(tcnt, 1.0f) : tsum;
    out[0] = (loss + track) * (1.0f / (float)NB);
}

// ---------------------------------------------------------------------------
extern "C" void kernel_launch(void* const* d_in, const int* in_sizes, int n_in,
                              void* d_out, int out_size, void* d_ws, size_t ws_size,
                              hipStream_t stream) {
    const float* seg  = (const float*)d_in[0];  // [B,5,H,W]
    const float* offp = (const float*)d_in[1];  // [B,2,H,W]
    const float* offs = (const float*)d_in[2];  // [B,2,H,W]
    const int* inst   = (const int*)d_in[3];
    const int* lab    = (const int*)d_in[4];
    const int* cent   = (const int*)d_in[5];
    float* out = (float*)d_out;

    char* ws = (char*)d_ws;
    float* stats   = (float*)(ws + 0);       // 4*16*16 f = 4 KB
    float* params  = (float*)(ws + 8192);    // 4*16*12 f = 3 KB
    float* bg      = (float*)(ws + 12288);   // 4*2 f
    float* seedSum = (float*)(ws + 12544);   // 64 f
    float* results = (float*)(ws + 13312);   // 96 f
    unsigned* errs = (unsigned*)(ws + 16384); // 96 * NPIX u32 = 96 MB

    // Zero the small accumulator region (4096 floats = 16 KB).
    k_zero<<<16, 256, 0, stream>>>((float*)ws, 4096);

    // Per-instance sums via WMMA.
    k_stats<<<dim3(64, NB), 256, 0, stream>>>(seg, offs, inst, lab, cent, stats, bg);
    k_finalize<<<1, 64, 0, stream>>>(stats, params);

    // Error key generation (96 arrays).
    k_errors<<<dim3(128, NB), 256, 0, stream>>>(seg, offp, inst, params, errs, seedSum);

    // Bitonic sort: local full sort of 2048-tiles, then global+local merges.
    dim3 gl(NPIX / TILE, NARR);                    // (128, 96)
    dim3 gg(NPIX / 512, NARR);                     // (512, 96), 256 thr
    k_sort_local<<<gl, 1024, 0, stream>>>(errs);
    for (unsigned k = 2 * TILE; k <= (unsigned)NPIX; k <<= 1) {
        for (unsigned j = k >> 1; j >= TILE; j >>= 1)
            k_sort_global<<<gg, 256, 0, stream>>>(errs, k, j);
        k_merge_local<<<gl, 1024, 0, stream>>>(errs, k);
    }

    k_lovasz<<<NARR, 256, 0, stream>>>(errs, params, results);
    k_final<<<1, 64, 0, stream>>>(params, seedSum, bg, results, out);
}